// DisGALayer_81527069213080
// MI455X (gfx1250) — compile-verified
//
#include <hip/hip_runtime.h>

typedef __attribute__((ext_vector_type(2))) float v2f;
typedef __attribute__((ext_vector_type(8))) float v8f;

#define IN_F 256
#define OUT_F 128

// ---------------------------------------------------------------------------
// K0: wa[0..255] = W @ a[:128];  wa[256..511] = W @ a[128:]
// ---------------------------------------------------------------------------
__global__ __launch_bounds__(256) void prep_wa(const float* __restrict__ W,
                                               const float* __restrict__ a,
                                               float* __restrict__ wa) {
    int k = threadIdx.x;              // 0..255
    float s0 = 0.f, s1 = 0.f;
    const float* wrow = W + (size_t)k * OUT_F;
    #pragma unroll 8
    for (int j = 0; j < OUT_F; ++j) {
        float w = wrow[j];
        s0 += w * a[j];
        s1 += w * a[OUT_F + j];
    }
    wa[k]       = s0;
    wa[256 + k] = s1;
}

// ---------------------------------------------------------------------------
// zero helper
// ---------------------------------------------------------------------------
__global__ __launch_bounds__(256) void zerok(float* __restrict__ p, long n) {
    long i = (long)blockIdx.x * 256 + threadIdx.x;
    if (i < n) p[i] = 0.f;
}

// ---------------------------------------------------------------------------
// K1: h_em = x @ W_em via V_WMMA_F32_16X16X4_F32, fused hs/hd matvecs.
// Block: 256 threads = 8 wave32s. Each block owns a 16-row strip of x,
// staged in LDS (contiguous 16*256 floats). Wave w computes the 16x16
// output tile at columns [16w, 16w+16).
// ---------------------------------------------------------------------------
__global__ __launch_bounds__(256) void gemm_fused(const float* __restrict__ x,
                                                  const float* __restrict__ Wem,
                                                  const float* __restrict__ wa,
                                                  float* __restrict__ h_em,
                                                  float* __restrict__ hs,
                                                  float* __restrict__ hd) {
    __shared__ float lds[16 * IN_F];  // 16 KB

    const int tid = threadIdx.x;
    const long rowBase = (long)blockIdx.x * 16;

    // cooperative load of the contiguous 16x256 strip (1024 float4s)
    const float4* xs4  = (const float4*)(x + rowBase * IN_F);
    float4*       lds4 = (float4*)lds;
    #pragma unroll
    for (int i = 0; i < 4; ++i) lds4[tid + 256 * i] = xs4[tid + 256 * i];
    __syncthreads();

    const int lane    = tid & 31;
    const int wid     = tid >> 5;       // 0..7 -> column tile
    const int colBase = wid * 16;
    const int m       = lane & 15;      // row (A) / col (B) within tile
    const int kh      = (lane >> 4) * 2; // K sub-offset: lanes 0-15 -> K0,1 ; 16-31 -> K2,3

    v8f c = {};
    #pragma unroll 4
    for (int k0 = 0; k0 < IN_F; k0 += 4) {
        v2f av, bv;
        // A fragment: row m of x-strip, K = k0+kh, k0+kh+1
        av.x = lds[m * IN_F + k0 + kh];
        av.y = lds[m * IN_F + k0 + kh + 1];
        // B fragment: col (colBase+m) of W_em, same K pair
        bv.x = Wem[(size_t)(k0 + kh) * OUT_F + colBase + m];
        bv.y = Wem[(size_t)(k0 + kh + 1) * OUT_F + colBase + m];
        c = __builtin_amdgcn_wmma_f32_16x16x4_f32(
                false, av, false, bv, (short)0, c, false, false);
    }

    // D layout: VGPR r -> M = r (lanes 0-15) / M = r+8 (lanes 16-31), N = lane&15
    const int rOff = (lane < 16) ? 0 : 8;
    #pragma unroll
    for (int r = 0; r < 8; ++r) {
        h_em[(rowBase + rOff + r) * OUT_F + colBase + m] = c[r];
    }

    // fused hs/hd: 32 threads, each a full 256-length dot from the LDS strip
    if (tid < 32) {
        int row  = tid & 15;
        int half = tid >> 4;           // 0 -> wa_src, 1 -> wa_dst
        const float* wv = wa + half * 256;
        float s = 0.f;
        #pragma unroll 8
        for (int k = 0; k < IN_F; ++k) s += lds[row * IN_F + k] * wv[k];
        if (half == 0) hs[rowBase + row] = s;
        else           hd[rowBase + row] = s;
    }
}

// ---------------------------------------------------------------------------
// K2: edge_e, v = exp(sigmoid(edge_e)), denom[src] += v
// (global max cancels in attention = v/denom -> omitted; sigmoid in (0,1))
// ---------------------------------------------------------------------------
__global__ __launch_bounds__(256) void edge_pass1(const int* __restrict__ ei,
                                                  const float* __restrict__ hs,
                                                  const float* __restrict__ hd,
                                                  float* __restrict__ edge_e_out,
                                                  float* __restrict__ v,
                                                  float* __restrict__ denom,
                                                  int E) {
    int e = blockIdx.x * 256 + threadIdx.x;
    if (e >= E) return;
    int s = ei[e];
    int d = ei[E + e];
    float ee = hs[s] + hd[d];
    float ob = 1.0f / (1.0f + __expf(-ee));
    float ve = __expf(ob);
    edge_e_out[e] = ee;
    v[e] = ve;
    __hip_atomic_fetch_add(&denom[s], ve, __ATOMIC_RELAXED, __HIP_MEMORY_SCOPE_AGENT);
}

// ---------------------------------------------------------------------------
// K3: h_prime[src] += (v/denom[src]) * h_em[dst]
// one wave32 per edge; lane L handles float4 at feature offset 4L.
// ---------------------------------------------------------------------------
__global__ __launch_bounds__(256) void edge_pass2(const int* __restrict__ ei,
                                                  const float* __restrict__ v,
                                                  const float* __restrict__ denom,
                                                  const float* __restrict__ h_em,
                                                  float* __restrict__ h_prime,
                                                  int E) {
    long gt   = (long)blockIdx.x * 256 + threadIdx.x;
    long e    = gt >> 5;
    int  lane = (int)(gt & 31);
    if (e >= E) return;
    int s = ei[e];
    int d = ei[E + e];
    float coeff = v[e] / denom[s];
    float4 hv = ((const float4*)(h_em + (size_t)d * OUT_F))[lane];
    float* outp = h_prime + (size_t)s * OUT_F + lane * 4;
    __hip_atomic_fetch_add(&outp[0], coeff * hv.x, __ATOMIC_RELAXED, __HIP_MEMORY_SCOPE_AGENT);
    __hip_atomic_fetch_add(&outp[1], coeff * hv.y, __ATOMIC_RELAXED, __HIP_MEMORY_SCOPE_AGENT);
    __hip_atomic_fetch_add(&outp[2], coeff * hv.z, __ATOMIC_RELAXED, __HIP_MEMORY_SCOPE_AGENT);
    __hip_atomic_fetch_add(&outp[3], coeff * hv.w, __ATOMIC_RELAXED, __HIP_MEMORY_SCOPE_AGENT);
}

// ---------------------------------------------------------------------------
extern "C" void kernel_launch(void* const* d_in, const int* in_sizes, int n_in,
                              void* d_out, int out_size, void* d_ws, size_t ws_size,
                              hipStream_t stream) {
    const float* x   = (const float*)d_in[0];   // (N, 256)
    const int*   ei  = (const int*)  d_in[1];   // (2, E)
    const float* W   = (const float*)d_in[2];   // (256, 128)
    const float* a   = (const float*)d_in[3];   // (256, 1)
    const float* Wem = (const float*)d_in[4];   // (256, 128)

    const int N = in_sizes[0] / IN_F;           // 100000
    const int E = in_sizes[1] / 2;              // 1600000

    // workspace layout (floats)
    float* ws    = (float*)d_ws;
    float* wa    = ws;                          // 512
    float* hs    = wa + 512;                    // N
    float* hd    = hs + N;                      // N
    float* denom = hd + N;                      // N
    float* vbuf  = denom + N;                   // E
    float* h_em  = vbuf + E;                    // N * 128

    float* h_prime    = (float*)d_out;          // N * 128
    float* edge_e_out = h_prime + (size_t)N * OUT_F;  // E

    // K0: wa = W @ [a_src | a_dst]
    prep_wa<<<1, 256, 0, stream>>>(W, a, wa);

    // zero denom and h_prime
    zerok<<<(N + 255) / 256, 256, 0, stream>>>(denom, N);
    zerok<<<(int)(((long)N * OUT_F + 255) / 256), 256, 0, stream>>>(h_prime, (long)N * OUT_F);

    // K1: h_em = x @ W_em (WMMA), hs/hd fused  (N is a multiple of 16)
    gemm_fused<<<N / 16, 256, 0, stream>>>(x, Wem, wa, h_em, hs, hd);

    // K2: edge scores + denom
    edge_pass1<<<(E + 255) / 256, 256, 0, stream>>>(ei, hs, hd, edge_e_out, vbuf, denom, E);

    // K3: weighted scatter-sum, one wave per edge
    long threads2 = (long)E * 32;
    edge_pass2<<<(int)((threads2 + 255) / 256), 256, 0, stream>>>(ei, vbuf, denom, h_em, h_prime, E);
}